// TriPlanePC2Encoder_85478439125638
// MI455X (gfx1250) — compile-verified
//
#include <hip/hip_runtime.h>
#include <hip/hip_bf16.h>
#include <stdint.h>

// ---------------------------------------------------------------------------
// Tri-plane point-cloud scatter-mean for MI455X (gfx1250).
//
// B=8, N=200000, C=32, R=128.  AI ~0.1 flop/byte -> memory/atomic bound; no
// matmul structure, so no WMMA.  CDNA5 leverage: the Tensor Data Mover loads
// each block's 256-point x 32-channel feature tile (32KB, stride N between
// channel rows) into LDS with ONE tensor_load_to_lds instruction (TENSORcnt),
// leaving the VMEM pipe free for the ~99 global_atomic_add per point that
// dominate.  The 50MB output lives in the 192MB L2, so the atomic RMW traffic
// stays on-chip.
// ---------------------------------------------------------------------------

#define NPTS 200000
#define NCH  32
#define RES  128
#define BINS (RES * RES)
#define BLK  256

typedef unsigned int v4u __attribute__((ext_vector_type(4)));
typedef int          v8i __attribute__((ext_vector_type(8)));
typedef int          v4i __attribute__((ext_vector_type(4)));

// Generic pointer to an LDS object: low 32 bits are the LDS byte offset
// (ISA 10.2: LDS aperture in addr[63:32]).
__device__ __forceinline__ uint32_t lds_off(const void* p) {
    return (uint32_t)(uintptr_t)p;
}

// ---------------------------------------------------------------------------
__global__ __launch_bounds__(BLK) void zero_kernel(float4* __restrict__ out4,
                                                   size_t n_out4,
                                                   uint4* __restrict__ cnt4,
                                                   size_t n_cnt4) {
    size_t i = (size_t)blockIdx.x * BLK + threadIdx.x;
    if (i < n_out4) out4[i] = make_float4(0.f, 0.f, 0.f, 0.f);
    if (i < n_cnt4) cnt4[i] = make_uint4(0u, 0u, 0u, 0u);
}

// ---------------------------------------------------------------------------
__global__ __launch_bounds__(BLK) void scatter_kernel(
    const float* __restrict__ xyz,    // (B, N, 3)
    const float* __restrict__ feat,   // (B, C, N)
    float* __restrict__ out,          // (B, 3, C, R, R)
    unsigned* __restrict__ cnt)       // (B, 3, R*R)
{
    // TDM destination tile: fbuf[ch][pt] (channel rows of 256 contiguous pts)
    __shared__ float fbuf[NCH * BLK];               // 32 KB

    const int tid  = threadIdx.x;
    const int b    = blockIdx.y;
    const int base = blockIdx.x * BLK;
    const int n    = base + tid;
    const bool active = (n < NPTS);
    const int nc = active ? n : (NPTS - 1);         // clamp tail (safe loads)

    // ---- TDM: one descriptor fetches the whole 256x32 feature tile -------
    if (tid < 32) {                                 // wave 0 only (EXEC is
        const uint64_t gaddr =                      //  ignored by TENSOR ops)
            (uint64_t)(uintptr_t)(feat + (size_t)b * NCH * NPTS + base);
        const uint32_t lds   = lds_off(&fbuf[0]);
        const uint32_t td0   = (uint32_t)(NPTS - base);   // OOB -> zero fill
        // D# group 0: count=1 | lds_addr | global_addr[56:0] | type=2
        v4u g0 = { 1u, lds, (uint32_t)gaddr,
                   (uint32_t)(((gaddr >> 32) & 0x1FFFFFFull) | 0x80000000u) };
        // D# group 1: data_size=4B; tensor (N-base) x 32; tile 256 x 32;
        //             dim0 stride = N elements; dim1 stride unused (2D).
        v8i g1 = { (int)0x00020000,
                   (int)((td0 & 0xFFFFu) << 16),
                   (int)(((td0 >> 16) & 0xFFFFu) | ((uint32_t)NCH << 16)),
                   (int)((uint32_t)BLK << 16),      // tensor_dim1[31:16]=0
                   (int)NCH,                        // tile_dim1=32, tile_dim2=0
                   (int)NPTS,                       // tensor_dim0_stride lo32
                   0, 0 };
        v4i gz4 = { 0, 0, 0, 0 };                   // 2D: groups 2/3 unused
        v8i gz8 = { 0, 0, 0, 0, 0, 0, 0, 0 };
        __builtin_amdgcn_tensor_load_to_lds(g0, g1, gz4, gz4, gz8, 0);
        __builtin_amdgcn_s_wait_tensorcnt(0);       // tile landed in LDS
    }
    __syncthreads();                                // publish to all 8 waves

    // ---- quantize point into the three planes (mirrors reference math) ---
    const float* p = xyz + ((size_t)b * NPTS + nc) * 3;
    const float eps = 1e-5f;
    int iq[3];
#pragma unroll
    for (int d = 0; d < 3; ++d) {
        float v = (p[d] * 0.5f) / (1.0f + eps) + 0.5f;
        v = fminf(fmaxf(v, 0.0f), 1.0f - eps);
        iq[d] = (int)(v * (float)RES);
    }
    const int lin0 = iq[0] + RES * iq[1];           // xy
    const int lin1 = iq[1] + RES * iq[2];           // yz
    const int lin2 = iq[0] + RES * iq[2];           // xz

    if (active) {
        unsigned* cb = cnt + (size_t)b * 3 * BINS;
        atomicAdd(cb + 0 * BINS + lin0, 1u);
        atomicAdd(cb + 1 * BINS + lin1, 1u);
        atomicAdd(cb + 2 * BINS + lin2, 1u);
    }

    // ---- channel loop: LDS-resident features -> 3 atomics per channel ----
    float* ob = out + (size_t)b * 3 * NCH * BINS;
    if (active) {
#pragma unroll 4
        for (int c = 0; c < NCH; ++c) {
            const float v = fbuf[c * BLK + tid];    // bank-conflict-free
            float* oc = ob + (size_t)c * BINS;
            atomicAdd(oc + 0 * (size_t)NCH * BINS + lin0, v);
            atomicAdd(oc + 1 * (size_t)NCH * BINS + lin1, v);
            atomicAdd(oc + 2 * (size_t)NCH * BINS + lin2, v);
        }
    }
}

// ---------------------------------------------------------------------------
__global__ __launch_bounds__(BLK) void normalize_kernel(
    float* __restrict__ out, const unsigned* __restrict__ cnt) {
    // one thread per 4 consecutive bins of one (b, plane); loop channels.
    const size_t i = (size_t)blockIdx.x * BLK + threadIdx.x; // [0, B*3*BINS/4)
    const uint4 c4 = ((const uint4*)cnt)[i];
    float4 inv;
    inv.x = 1.0f / fmaxf((float)c4.x, 1.0f);
    inv.y = 1.0f / fmaxf((float)c4.y, 1.0f);
    inv.z = 1.0f / fmaxf((float)c4.z, 1.0f);
    inv.w = 1.0f / fmaxf((float)c4.w, 1.0f);
    const size_t bin0 = i * 4;
    const size_t row  = bin0 / BINS;                // (b*3 + plane)
    float4* o = (float4*)(out + row * (size_t)NCH * BINS + (bin0 % BINS));
#pragma unroll
    for (int ch = 0; ch < NCH; ++ch) {
        float4 v = o[(size_t)ch * (BINS / 4)];
        v.x *= inv.x; v.y *= inv.y; v.z *= inv.z; v.w *= inv.w;
        o[(size_t)ch * (BINS / 4)] = v;
    }
}

// ---------------------------------------------------------------------------
extern "C" void kernel_launch(void* const* d_in, const int* in_sizes, int n_in,
                              void* d_out, int out_size, void* d_ws,
                              size_t ws_size, hipStream_t stream) {
    const float* xyz  = (const float*)d_in[0];   // (B, N, 3) f32
    const float* feat = (const float*)d_in[1];   // (B, C, N) f32
    float* out = (float*)d_out;                  // (B, 3, C, R, R) f32
    unsigned* cnt = (unsigned*)d_ws;             // (B, 3, R*R) u32 scratch

    const int B = in_sizes[0] / (NPTS * 3);      // = 8
    const size_t n_out4 = (size_t)out_size / 4;  // divisible (C*R*R mult of 4)
    const size_t n_cnt  = (size_t)B * 3 * BINS;
    const size_t n_cnt4 = n_cnt / 4;

    // Phase 1: zero accumulators (harness poisons d_out / d_ws with 0xAA).
    const size_t nz = (n_out4 > n_cnt4) ? n_out4 : n_cnt4;
    zero_kernel<<<(unsigned)((nz + BLK - 1) / BLK), BLK, 0, stream>>>(
        (float4*)out, n_out4, (uint4*)cnt, n_cnt4);

    // Phase 2: TDM-staged scatter-add.
    dim3 grid((NPTS + BLK - 1) / BLK, B);
    scatter_kernel<<<grid, BLK, 0, stream>>>(xyz, feat, out, cnt);

    // Phase 3: divide by counts (empty bins stay 0 via max(cnt,1)).
    normalize_kernel<<<(unsigned)(n_cnt4 / BLK), BLK, 0, stream>>>(out, cnt);
}